// JpegCompression_27307402068560
// MI455X (gfx1250) — compile-verified
//
#include <hip/hip_runtime.h>
#include <math.h>

// ---------------------------------------------------------------------------
// Differentiable JPEG (quality=50) fused kernel for MI455X (gfx1250, wave32).
// Memory-bound: ~201MB traffic -> ~8.6us at 23.3 TB/s. All DCT/IDCT 8x8
// transforms are batched 4-at-a-time into 16x16 WMMA tiles using a
// block-diagonal DCT basis, computed with chained v_wmma_f32_16x16x4_f32.
// The basis fragments are hoisted into registers (one v2f[4] set serves the
// A-operand of stages 1/4 and the B-operand of stages 2/3) so the inner
// stages contain only unconditional ds_load/ds_store + wmma.
// ---------------------------------------------------------------------------

typedef __attribute__((ext_vector_type(2))) float v2f;
typedef __attribute__((ext_vector_type(8))) float v8f;

// Standard JPEG quantization tables (quality factor 1.0 at QUALITY=50)
__constant__ float cQY[64] = {
    16, 11, 10, 16, 24, 40, 51, 61,
    12, 12, 14, 19, 26, 58, 60, 55,
    14, 13, 16, 24, 40, 57, 69, 56,
    14, 17, 22, 29, 51, 87, 80, 62,
    18, 22, 37, 56, 68, 109, 103, 77,
    24, 35, 55, 64, 81, 104, 113, 92,
    49, 64, 78, 87, 103, 121, 120, 101,
    72, 92, 95, 98, 112, 100, 103, 99};
__constant__ float cQC[64] = {
    17, 18, 24, 47, 99, 99, 99, 99,
    18, 21, 26, 66, 99, 99, 99, 99,
    24, 26, 56, 99, 99, 99, 99, 99,
    47, 66, 99, 99, 99, 99, 99, 99,
    99, 99, 99, 99, 99, 99, 99, 99,
    99, 99, 99, 99, 99, 99, 99, 99,
    99, 99, 99, 99, 99, 99, 99, 99,
    99, 99, 99, 99, 99, 99, 99, 99};

__device__ __forceinline__ float alphaf(int u) {
    return (u == 0) ? 0.70710678118654752f : 1.0f;
}

__device__ __forceinline__ v8f wmma4(v2f a, v2f b, v8f c) {
    // D(16x16,f32) += A(16x4,f32) * B(4x16,f32)
    return __builtin_amdgcn_wmma_f32_16x16x4_f32(
        /*neg_a=*/false, a, /*neg_b=*/false, b,
        /*c_mod=*/(short)0, c, /*reuse_a=*/false, /*reuse_b=*/false);
}

#define YSTR 33  // padded luma row stride (floats) to avoid LDS bank conflicts

__global__ __launch_bounds__(192) void jpeg50_kernel(const float* __restrict__ x,
                                                     float* __restrict__ out) {
    __shared__ float sA16[16 * 16];   // full block-diagonal 16x16 DCT basis
    __shared__ float sY[32 * YSTR];   // luma plane of this 32x32 region (padded)
    __shared__ float sCb[32 * 32];    // full-res chroma (pre-subsample)
    __shared__ float sCr[32 * 32];
    __shared__ float sCbs[16 * 16];   // subsampled chroma tiles (in-place)
    __shared__ float sCrs[16 * 16];

    const int tid  = threadIdx.x;
    const int lane = tid & 31;
    const int wave = tid >> 5;       // 6 waves: 0-3 luma tiles, 4 Cb, 5 Cr
    const int half = lane >> 4;      // lane group (0: lanes 0-15, 1: 16-31)
    const int lm   = lane & 15;

    const int blk = blockIdx.x;
    const int b   = blk >> 8;        // image index (256 regions per image)
    const int reg = blk & 255;
    const int h0  = (reg >> 4) << 5; // 32x32 region origin
    const int w0  = (reg & 15) << 5;

    // Block-diagonal basis: A16[r][c] = cos((2r+1)c*pi/16) within each 8x8
    // diagonal block, 0 elsewhere (zeros materialized -> no branches later).
    for (int i = tid; i < 256; i += 192) {
        int r = i >> 4, c = i & 15;
        float v = ((r ^ c) & 8) ? 0.0f
                : cosf(0.19634954084936207f * (float)((2 * (r & 7) + 1) * (c & 7)));
        sA16[i] = v;
    }

    const size_t planeStride = (size_t)512 * 512;
    const float* pr = x + ((size_t)b * 3 + 0) * planeStride;
    const float* pg = x + ((size_t)b * 3 + 1) * planeStride;
    const float* pb = x + ((size_t)b * 3 + 2) * planeStride;

    // ---- Phase 1a: vectorized load (global_load_b128), RGB -> YCbCr ----
    for (int i = tid; i < 256; i += 192) {
        int py = i >> 3;
        int px = (i & 7) << 2;
        size_t off = (size_t)(h0 + py) * 512 + (size_t)(w0 + px);
        float4 rv = *(const float4*)(pr + off);
        float4 gv = *(const float4*)(pg + off);
        float4 bv = *(const float4*)(pb + off);
        float rr[4] = {rv.x, rv.y, rv.z, rv.w};
        float gg[4] = {gv.x, gv.y, gv.z, gv.w};
        float bb[4] = {bv.x, bv.y, bv.z, bv.w};
#pragma unroll
        for (int k = 0; k < 4; ++k) {
            float R = fminf(fmaxf(rr[k], 0.f), 1.f) * 255.f;
            float G = fminf(fmaxf(gg[k], 0.f), 1.f) * 255.f;
            float B = fminf(fmaxf(bb[k], 0.f), 1.f) * 255.f;
            float Y  = 0.299f * R + 0.587f * G + 0.114f * B;
            float Cb = -0.168736f * R - 0.331264f * G + 0.5f * B + 128.f;
            float Cr = 0.5f * R - 0.418688f * G - 0.081312f * B + 128.f;
            sY[py * YSTR + px + k] = Y;
            sCb[py * 32 + px + k] = Cb;
            sCr[py * 32 + px + k] = Cr;
        }
    }
    __syncthreads();

    // Per-lane basis fragments, hoisted to registers. For K-chunk kc the lane
    // holds A16[k0][lm], A16[k0+1][lm] with k0 = 4*kc + 2*half. These serve
    // as A-operand (= A16^T slice) in stages 1/4 AND B-operand (= A16 row
    // slice) in stages 2/3.
    v2f aF[4];
#pragma unroll
    for (int kc = 0; kc < 4; ++kc) {
        int k0 = kc * 4 + 2 * half;
        aF[kc].x = sA16[k0 * 16 + lm];
        aF[kc].y = sA16[(k0 + 1) * 16 + lm];
    }

    // ---- Phase 1b: 4:2:0 chroma subsample (2x2 mean) ----
    for (int i = tid; i < 256; i += 192) {
        int sy = i >> 4, sx = i & 15;
        int base = (sy * 2) * 32 + sx * 2;
        sCbs[i] = 0.25f * (sCb[base] + sCb[base + 1] + sCb[base + 32] + sCb[base + 33]);
        sCrs[i] = 0.25f * (sCr[base] + sCr[base + 1] + sCr[base + 32] + sCr[base + 33]);
    }
    __syncthreads();

    // ---- Tile assignment (per wave): pointer + stride + quant table ----
    float* tp; int ts; const float* qtab;
    if (wave < 4) {
        tp = sY + ((wave >> 1) * 16) * YSTR + (wave & 1) * 16;  // 2x2 luma tiles
        ts = YSTR; qtab = cQY;
    } else {
        tp = (wave == 4) ? sCbs : sCrs;
        ts = 16; qtab = cQC;
    }
    const float factor = 1.0f;  // QUALITY=50 -> (200 - 2*50)/100

    // ---- Stage 1: T = A16^T * (X - 128)   (4 chained K=4 WMMAs) ----
    {
        v8f acc = {};
#pragma unroll
        for (int kc = 0; kc < 4; ++kc) {
            int k0 = kc * 4 + 2 * half;
            v2f bm;
            bm.x = tp[k0 * ts + lm] - 128.f;
            bm.y = tp[(k0 + 1) * ts + lm] - 128.f;
            acc = wmma4(aF[kc], bm, acc);
        }
#pragma unroll
        for (int v = 0; v < 8; ++v) tp[(v + 8 * half) * ts + lm] = acc[v];
    }
    __syncthreads();

    // ---- Stage 2: D = T * A16, then quantize/dequantize elementwise ----
    {
        v8f acc = {};
#pragma unroll
        for (int kc = 0; kc < 4; ++kc) {
            int k0 = kc * 4 + 2 * half;
            v2f am;
            am.x = tp[lm * ts + k0];       // T as A-operand (consecutive -> b64)
            am.y = tp[lm * ts + k0 + 1];
            acc = wmma4(am, aF[kc], acc);
        }
#pragma unroll
        for (int v = 0; v < 8; ++v) {
            int u  = v + 8 * half;         // freq row
            int vc = lm;                   // freq col
            int iu = u & 7, iv = vc & 7;
            float al  = alphaf(iu) * alphaf(iv);
            float dct = 0.25f * al * acc[v];                 // SCALE * (A^T X A)
            float q   = qtab[iu * 8 + iv] * factor;
            float t   = dct / q;
            float r   = rintf(t);                            // round-half-even
            float d   = t - r;
            float soft = r + d * d * d;                      // diff_round
            float deq  = soft * q;
            tp[u * ts + vc] = deq * al;                      // pre-scale ALPHA2
        }
    }
    __syncthreads();

    // ---- Stage 3: M = Deq' * A16 ----
    {
        v8f acc = {};
#pragma unroll
        for (int kc = 0; kc < 4; ++kc) {
            int k0 = kc * 4 + 2 * half;
            v2f am;
            am.x = tp[lm * ts + k0];
            am.y = tp[lm * ts + k0 + 1];
            acc = wmma4(am, aF[kc], acc);
        }
#pragma unroll
        for (int v = 0; v < 8; ++v) tp[(v + 8 * half) * ts + lm] = acc[v];
    }
    __syncthreads();

    // ---- Stage 4: recon = 0.25 * (A16^T * M) + 128 ----
    {
        v8f acc = {};
#pragma unroll
        for (int kc = 0; kc < 4; ++kc) {
            int k0 = kc * 4 + 2 * half;
            v2f bm;
            bm.x = tp[k0 * ts + lm];
            bm.y = tp[(k0 + 1) * ts + lm];
            acc = wmma4(aF[kc], bm, acc);
        }
#pragma unroll
        for (int v = 0; v < 8; ++v)
            tp[(v + 8 * half) * ts + lm] = 0.25f * acc[v] + 128.f;
    }
    __syncthreads();

    // ---- Phase 3: upsample chroma, YCbCr -> RGB, vectorized store ----
    float* qr = out + ((size_t)b * 3 + 0) * planeStride;
    float* qg = out + ((size_t)b * 3 + 1) * planeStride;
    float* qb = out + ((size_t)b * 3 + 2) * planeStride;
    for (int i = tid; i < 256; i += 192) {
        int py = i >> 3;
        int px = (i & 7) << 2;
        float ro[4], go[4], bo[4];
#pragma unroll
        for (int k = 0; k < 4; ++k) {
            float Y  = sY[py * YSTR + px + k];
            int ci   = (py >> 1) * 16 + ((px + k) >> 1);
            float Cb = sCbs[ci] - 128.f;
            float Cr = sCrs[ci] - 128.f;
            float R = Y + 1.402f * Cr;
            float G = Y - 0.344136f * Cb - 0.714136f * Cr;
            float B = Y + 1.772f * Cb;
            ro[k] = fminf(fmaxf(R, 0.f), 255.f) * (1.0f / 255.f);
            go[k] = fminf(fmaxf(G, 0.f), 255.f) * (1.0f / 255.f);
            bo[k] = fminf(fmaxf(B, 0.f), 255.f) * (1.0f / 255.f);
        }
        size_t off = (size_t)(h0 + py) * 512 + (size_t)(w0 + px);
        *(float4*)(qr + off) = make_float4(ro[0], ro[1], ro[2], ro[3]);
        *(float4*)(qg + off) = make_float4(go[0], go[1], go[2], go[3]);
        *(float4*)(qb + off) = make_float4(bo[0], bo[1], bo[2], bo[3]);
    }
}

extern "C" void kernel_launch(void* const* d_in, const int* in_sizes, int n_in,
                              void* d_out, int out_size, void* d_ws, size_t ws_size,
                              hipStream_t stream) {
    (void)in_sizes; (void)n_in; (void)d_ws; (void)ws_size; (void)out_size;
    const float* x = (const float*)d_in[0];
    float* out = (float*)d_out;
    // 32 images * (512/32)^2 = 8192 regions; 192 threads = 6 wave32 waves.
    jpeg50_kernel<<<dim3(32 * 256), dim3(192), 0, stream>>>(x, out);
}